// RelationAwareLinear_5506148074156
// MI455X (gfx1250) — compile-verified
//
#include <hip/hip_runtime.h>

// Problem constants (from reference): N=4096, R=16, DIN=256, DOUT=256
constexpr int NROWS = 4096;
constexpr int NREL  = 16;
constexpr int DIN   = 256;
constexpr int DOUT  = 256;
constexpr int CHUNK = 16;                 // rows per GEMM tile (WMMA M)
constexpr int NCHUNK = NROWS / CHUNK;     // 256
constexpr int XS_STRIDE = 260;            // dword stride: 260 % 64 = 4 -> 16 rows hit 16 distinct banks; 16B aligned

typedef __attribute__((ext_vector_type(2))) float v2f;
typedef __attribute__((ext_vector_type(8))) float v8f;

// ---------------- bucketing prep ----------------

__global__ void rel_zero_kernel(int* counts, int* cursor) {
    int t = threadIdx.x;
    if (t < NREL) counts[t] = 0;
    else if (t < 2 * NREL) cursor[t - NREL] = 0;
}

__global__ void rel_count_kernel(const int* __restrict__ rel, int* counts) {
    int n = blockIdx.x * blockDim.x + threadIdx.x;
    if (n < NROWS) atomicAdd(&counts[rel[n]], 1);
}

__global__ void rel_scan_kernel(const int* __restrict__ counts, int* offsets) {
    if (threadIdx.x == 0) {
        int s = 0;
        for (int r = 0; r < NREL; ++r) { offsets[r] = s; s += counts[r]; }
    }
}

__global__ void rel_scatter_kernel(const int* __restrict__ rel,
                                   const int* __restrict__ offsets,
                                   int* cursor, int* perm) {
    int n = blockIdx.x * blockDim.x + threadIdx.x;
    if (n < NROWS) {
        int r = rel[n];
        int pos = offsets[r] + atomicAdd(&cursor[r], 1);
        perm[pos] = n;   // permuted order is sorted by relation
    }
}

// ---------------- bucketed GEMM with f32 WMMA ----------------
// grid = NCHUNK, block = 512 (16 wave32s). One block owns one 16-row gathered
// chunk (staged once into LDS) and all 256 output columns; wave w computes
// cols [w*16, w*16+16) as one 16x16 f32 WMMA tile over K=256.

__global__ __launch_bounds__(512) void rel_gemm_wmma_kernel(
    const float* __restrict__ x,
    const int*   __restrict__ rel,
    const float* __restrict__ w,
    const float* __restrict__ bias,
    const int*   __restrict__ perm,
    float*       __restrict__ out) {

    __shared__ float xs[CHUNK * XS_STRIDE];   // 16 x 256 f32, bank-padded
    __shared__ int   permRow[CHUNK];
    __shared__ int   relRow[CHUNK];

    const int tid  = threadIdx.x;
    const int wave = tid >> 5;     // 0..15
    const int lane = tid & 31;
    const int half = lane >> 4;    // 0 or 1
    const int idx  = lane & 15;    // M (for A) / N (for B,C,D)

    const int chunk = blockIdx.x;
    const int col   = wave * 16 + idx;        // 0..255

    if (tid < CHUNK) {
        int n = perm[chunk * CHUNK + tid];
        permRow[tid] = n;
        relRow[tid]  = rel[n];
    }
    __syncthreads();

    // Stage the 16 gathered x-rows into LDS (float4 cooperative copy).
    for (int i = tid; i < CHUNK * (DIN / 4); i += 512) {
        int row = i >> 6;          // / (DIN/4)
        int c4  = i & 63;
        float4 v = reinterpret_cast<const float4*>(x + (size_t)permRow[row] * DIN)[c4];
        *reinterpret_cast<float4*>(&xs[row * XS_STRIDE + c4 * 4]) = v;
    }
    __syncthreads();

    // Permuted rows are sorted by relation -> chunk is a few contiguous runs
    // (usually exactly one). Run loop is wave-uniform; WMMA always executes
    // with EXEC all-1s, only loads/stores are lane-predicated.
    const float* xrow = &xs[idx * XS_STRIDE];

    int m = 0;
    while (m < CHUNK) {
        const int r = relRow[m];
        int e = m + 1;
        while (e < CHUNK && relRow[e] == r) ++e;

        const float* wr = w + (size_t)r * DIN * DOUT;
        v8f acc = {};

        if (m == 0 && e == CHUNK) {
            // ---- fast path: homogeneous chunk, no A masking ----
            #pragma unroll 8
            for (int k0 = 0; k0 < DIN; k0 += 4) {
                const int ka = k0 + 2 * half;             // this half's K pair
                v2f a = *reinterpret_cast<const v2f*>(xrow + ka);  // 8B-aligned LDS
                v2f b;
                b.x = wr[(size_t)ka * DOUT + col];
                b.y = wr[(size_t)(ka + 1) * DOUT + col];
                acc = __builtin_amdgcn_wmma_f32_16x16x4_f32(
                    false, a, false, b, (short)0, acc, false, false);
            }
            const float bv = bias[r * DOUT + col];
            #pragma unroll
            for (int v = 0; v < 8; ++v) {
                const int M = v + 8 * half;               // C/D row held in VGPR v
                out[(size_t)permRow[M] * DOUT + col] = acc[v] + bv;
            }
        } else {
            // ---- boundary path: mask A rows outside [m, e) ----
            const bool aValid = (idx >= m) & (idx < e);
            #pragma unroll 8
            for (int k0 = 0; k0 < DIN; k0 += 4) {
                const int ka = k0 + 2 * half;
                v2f av = *reinterpret_cast<const v2f*>(xrow + ka);
                v2f a;
                a.x = aValid ? av.x : 0.0f;
                a.y = aValid ? av.y : 0.0f;
                v2f b;
                b.x = wr[(size_t)ka * DOUT + col];
                b.y = wr[(size_t)(ka + 1) * DOUT + col];
                acc = __builtin_amdgcn_wmma_f32_16x16x4_f32(
                    false, a, false, b, (short)0, acc, false, false);
            }
            const float bv = bias[r * DOUT + col];
            #pragma unroll
            for (int v = 0; v < 8; ++v) {
                const int M = v + 8 * half;
                if (M >= m && M < e) {
                    out[(size_t)permRow[M] * DOUT + col] = acc[v] + bv;
                }
            }
        }
        m = e;
    }
}

// ---------------- launch ----------------

extern "C" void kernel_launch(void* const* d_in, const int* in_sizes, int n_in,
                              void* d_out, int out_size, void* d_ws, size_t ws_size,
                              hipStream_t stream) {
    const float* x    = (const float*)d_in[0];   // [N, DIN]
    const int*   rel  = (const int*)  d_in[1];   // [N]
    const float* w    = (const float*)d_in[2];   // [R, DIN, DOUT]
    const float* bias = (const float*)d_in[3];   // [R, DOUT]
    float*       out  = (float*)d_out;           // [N, DOUT]

    int* ws      = (int*)d_ws;
    int* counts  = ws;             // 16
    int* offsets = ws + 16;        // 16
    int* cursor  = ws + 32;        // 16
    int* perm    = ws + 48;        // 4096

    rel_zero_kernel<<<1, 32, 0, stream>>>(counts, cursor);
    rel_count_kernel<<<NROWS / 256, 256, 0, stream>>>(rel, counts);
    rel_scan_kernel<<<1, 32, 0, stream>>>(counts, offsets);
    rel_scatter_kernel<<<NROWS / 256, 256, 0, stream>>>(rel, offsets, cursor, perm);

    rel_gemm_wmma_kernel<<<NCHUNK, 512, 0, stream>>>(x, rel, w, bias, perm, out);
}